// Encoder_20426864460432
// MI455X (gfx1250) — compile-verified
//
#include <hip/hip_runtime.h>
#include <hip/hip_bf16.h>

typedef __attribute__((ext_vector_type(2))) float v2f;
typedef __attribute__((ext_vector_type(8))) float v8f;

#define DIM        256
#define DIM2       512
#define EMB_COLS   255
#define LDS_STRIDE 260   // 256 + 4 pad: row stride mod 64 banks = 4 -> conflict-free

// ---------------------------------------------------------------------------
// Fused embedding-gather + mask + GEMM head:  params = (mask*[charge|emb]) @ W + b
// One block = 16 rows (one M-tile slab), 8 waves x 64 columns each.
// fp32 WMMA (V_WMMA_F32_16X16X4_F32) -> exact fp32 math, K loop of 64 steps.
// ---------------------------------------------------------------------------
__global__ __launch_bounds__(256) void encoder_gemm_kernel(
    const int*   __restrict__ categories,
    const float* __restrict__ charges,
    const float* __restrict__ node_mask,
    const float* __restrict__ emb_table,
    const float* __restrict__ W,       // [DIM][DIM2] row-major
    const float* __restrict__ b_lin,   // [DIM2]
    float*       __restrict__ params)  // [nnodes][DIM2]
{
    __shared__ float hs[16 * LDS_STRIDE];

    const int tid = threadIdx.x;
    const int m0  = blockIdx.x * 16;

    // ---- Build the 16x256 h slab in LDS (charge col 0, embedding cols 1..255)
    for (int idx = tid; idx < 16 * DIM; idx += 256) {
        const int r    = idx >> 8;       // DIM == 256
        const int c    = idx & (DIM - 1);
        const int node = m0 + r;
        const float nm = node_mask[node];
        float v;
        if (c == 0) {
            v = charges[node];
        } else {
            const int cat = categories[node];
            v = emb_table[cat * EMB_COLS + (c - 1)];
        }
        hs[r * LDS_STRIDE + c] = v * nm;
    }
    __syncthreads();

    const int wave = tid >> 5;           // 0..7
    const int lane = tid & 31;
    const int n0   = wave * 64;          // 4 N-tiles of 16 per wave

    // A fragment (16x4 f32): lane<16 -> K=k+0,k+1 ; lane>=16 -> K=k+2,k+3 of row lane&15
    const int arow     = lane & 15;
    const int acol_off = (lane >> 4) * 2;
    // B fragment (4x16 f32): lane<16 -> rows K=k+0,k+1 ; lane>=16 -> rows K=k+2,k+3, col lane&15
    const int bn       = lane & 15;
    const int brow_off = (lane >> 4) * 2;

    v8f acc[4];
    #pragma unroll
    for (int t = 0; t < 4; ++t) {
        const float bv = b_lin[n0 + t * 16 + bn];   // C layout: N = lane&15 in every VGPR
        #pragma unroll
        for (int j = 0; j < 8; ++j) acc[t][j] = bv;
    }

    for (int k = 0; k < DIM; k += 4) {
        // one 64-bit LDS read per lane (8B aligned: row stride 1040B, col even)
        const v2f a = *(const v2f*)&hs[arow * LDS_STRIDE + k + acol_off];
        const float* wbase = &W[(k + brow_off) * DIM2 + n0 + bn];
        #pragma unroll
        for (int t = 0; t < 4; ++t) {
            v2f b;
            b.x = wbase[t * 16];
            b.y = wbase[t * 16 + DIM2];
            acc[t] = __builtin_amdgcn_wmma_f32_16x16x4_f32(
                false, a, false, b, (short)0, acc[t], false, false);
        }
    }

    // D layout: VGPR j -> M=j (lanes 0-15) / M=j+8 (lanes 16-31); N = lane&15.
    // Streaming output (64 MB total) -> non-temporal to keep W/x hot in L2.
    const int mbase = m0 + (lane >> 4) * 8;
    #pragma unroll
    for (int t = 0; t < 4; ++t) {
        float* out = &params[(size_t)mbase * DIM2 + n0 + t * 16 + bn];
        #pragma unroll
        for (int j = 0; j < 8; ++j) {
            __builtin_nontemporal_store(acc[t][j], out + (size_t)j * DIM2);
        }
    }
}

// ---------------------------------------------------------------------------
// coord2diff edge pass: bandwidth bound (33.5 MB idx in, 67 MB out).
// x table (384 KB) is cache-resident; indices and results streamed NT.
// ---------------------------------------------------------------------------
__global__ __launch_bounds__(256) void coord2diff_kernel(
    const int*   __restrict__ edges,   // [2][E] flattened
    const float* __restrict__ x,       // [nnodes][3]
    float*       __restrict__ radial,  // [E]
    float*       __restrict__ cd,      // [E][3]
    int E)
{
    const int e = blockIdx.x * 256 + threadIdx.x;
    if (e >= E) return;

    const int r = __builtin_nontemporal_load(&edges[e]);
    const int c = __builtin_nontemporal_load(&edges[E + e]);

    const float dx = x[r * 3 + 0] - x[c * 3 + 0];
    const float dy = x[r * 3 + 1] - x[c * 3 + 1];
    const float dz = x[r * 3 + 2] - x[c * 3 + 2];

    const float rad  = dx * dx + dy * dy + dz * dz;
    const float norm = sqrtf(rad + 1e-8f);
    const float inv  = 1.0f / (norm + 1.0f);   // NORM_CONSTANT = 1.0

    __builtin_nontemporal_store(rad,      &radial[e]);
    __builtin_nontemporal_store(dx * inv, &cd[3 * e + 0]);
    __builtin_nontemporal_store(dy * inv, &cd[3 * e + 1]);
    __builtin_nontemporal_store(dz * inv, &cd[3 * e + 2]);
}

extern "C" void kernel_launch(void* const* d_in, const int* in_sizes, int n_in,
                              void* d_out, int out_size, void* d_ws, size_t ws_size,
                              hipStream_t stream) {
    (void)n_in; (void)out_size; (void)d_ws; (void)ws_size;

    const float* x          = (const float*)d_in[0];   // [B,N,3]
    const int*   categories = (const int*)  d_in[1];   // [B,N]
    const float* charges    = (const float*)d_in[2];   // [B,N,1]
    const int*   edges      = (const int*)  d_in[3];   // [2,E]
    const float* node_mask  = (const float*)d_in[4];   // [B,N,1]
    /* d_in[5] edge_mask: unused by the reference computation */
    const float* emb_table  = (const float*)d_in[6];   // [MAX_Z, 255]
    const float* W          = (const float*)d_in[7];   // [256, 512]
    const float* b_lin      = (const float*)d_in[8];   // [512]

    const int nnodes = in_sizes[1];        // B*N = 32768
    const int E      = in_sizes[3] / 2;    // 4,194,304

    float* params = (float*)d_out;                         // [nnodes, 512]
    float* radial = params + (size_t)nnodes * DIM2;        // [E]
    float* cd     = radial + (size_t)E;                    // [E, 3]

    encoder_gemm_kernel<<<nnodes / 16, 256, 0, stream>>>(
        categories, charges, node_mask, emb_table, W, b_lin, params);

    coord2diff_kernel<<<(E + 255) / 256, 256, 0, stream>>>(
        edges, x, radial, cd, E);
}